// Im2Col_Conv2d_6854767804820
// MI455X (gfx1250) — compile-verified
//
#include <hip/hip_runtime.h>

typedef float v2f __attribute__((ext_vector_type(2)));
typedef float v8f __attribute__((ext_vector_type(8)));
typedef int   v4i __attribute__((vector_size(16)));   // matches builtin's int4 param

#if defined(__has_builtin)
#  if __has_builtin(__builtin_amdgcn_global_load_async_to_lds_b128)
#    define USE_ASYNC_LDS 1
#  endif
#endif
#ifndef USE_ASYNC_LDS
#  define USE_ASYNC_LDS 0
#endif

typedef __attribute__((address_space(1))) v4i* gv4p;  // global int4*
typedef __attribute__((address_space(3))) v4i* lv4p;  // LDS int4*

namespace {
constexpr int Hin = 256, Win = 256, Cin = 8, Mout = 8;
constexpr int Hout = 254, Wout = 254;
constexpr int RWG   = 8;            // output rows per workgroup
constexpr int SROWS = RWG + 2;      // staged input rows (10)
constexpr int KTOT  = 72;           // Cin*3*3
constexpr int NSTEP = 18;           // wmma steps of K=4 (2 k's per half-wave)
constexpr int CSTR  = 260;          // padded col stride: 16-bank skew, absorbs col 256..257 overreads
constexpr int RSTR  = Cin * CSTR;   // 2080 elements per staged row
constexpr size_t LDS_BYTES = (size_t)SROWS * RSTR * sizeof(float); // 83,200 B
}

// K permutation: idx = 2*st (+1) in [0,36): c=idx/9 (0..3), r,s from remainder;
// the hi half-wave's +4 channels live in the per-lane base register.
__host__ __device__ constexpr int ldsOff(int idx) {
  int c = idx / 9, rem = idx % 9, r = rem / 3, s = rem % 3;
  return r * RSTR + c * CSTR + s;
}

__global__ __launch_bounds__(256) void conv3x3_wmma_f32(
    const float* __restrict__ in, const float* __restrict__ flt,
    float* __restrict__ out) {
  extern __shared__ float lds[];    // [SROWS][Cin][CSTR]

  const int tid  = threadIdx.x;
  const int lane = tid & 31;
  const int wid  = tid >> 5;        // 8 waves, one output row each
  const int n    = blockIdx.x >> 5;
  const int blk  = blockIdx.x & 31;
  const int i0   = blk * RWG;

  // ---- stage input rows [i0, i0+9] x 8 channels into LDS (division-free) ----
  const float* inN = in + (size_t)n * Cin * Hin * Win;
  {
    const int c   = tid >> 5;       // one 32-thread group per channel
    const int t32 = tid & 31;
    #pragma unroll
    for (int it = 0; it < 20; ++it) {
      int idx2 = t32 + it * 32;     // 0..639 float4 tiles within channel
      int rr   = idx2 >> 6;         // staged row 0..9
      int c4   = (idx2 & 63) * 4;   // column (floats)
      int grow = i0 + rr; if (grow > Hin - 1) grow = Hin - 1;
      const float* gsrc = inN + (size_t)(c * Hin + grow) * Win + c4;
      float* ldst = &lds[rr * RSTR + c * CSTR + c4];
#if USE_ASYNC_LDS
      __builtin_amdgcn_global_load_async_to_lds_b128(
          (gv4p)(void*)(const_cast<float*>(gsrc)), (lv4p)(void*)ldst, 0, 0);
#else
      *(float4*)ldst = *(const float4*)gsrc;
#endif
    }
  }
#if USE_ASYNC_LDS
#  if defined(__has_builtin) && __has_builtin(__builtin_amdgcn_s_wait_asynccnt)
  __builtin_amdgcn_s_wait_asynccnt(0);
#  else
  asm volatile("s_wait_asynccnt 0" ::: "memory");
#  endif
#endif
  __syncthreads();

  const int m  = lane & 15;   // A row: output channel (8 real + 8 pad)
  const int hi = lane >> 4;   // half-wave: K slots {0,1} = c 0..3, {2,3} = c 4..7
  const int p  = lane & 15;   // B column: output position within tile

  // ---- A fragments (filter) resident in VGPRs; contiguous thanks to c-split ----
  const float* fbase = flt + m * KTOT + hi * 36;
  v2f a[NSTEP];
  #pragma unroll
  for (int st = 0; st < NSTEP; ++st) {
    v2f av = {0.f, 0.f};
    if (m < Mout) av = *(const v2f*)(fbase + 2 * st);
    a[st] = av;
  }

  const int i = i0 + wid;
  if (i >= Hout) return;      // uniform per-wave exit: EXEC all-ones for WMMA

  // single per-lane LDS base; all per-step/tile variation is compile-time immediates
  const unsigned ldsIdx = (unsigned)(wid * RSTR + hi * (4 * CSTR) + p);
  char* outb = (char*)(out + ((size_t)n * Mout * Hout + i) * Wout + p);

  #pragma unroll
  for (int j0 = 0; j0 < 256; j0 += 16) {    // 16 column tiles
    v8f acc0 = {}, acc1 = {};               // two chains for ILP
    #pragma unroll
    for (int st = 0; st < NSTEP; st += 2) {
      v2f b0, b1;
      b0.x = lds[ldsIdx + j0 + ldsOff(2 * st)];
      b0.y = lds[ldsIdx + j0 + ldsOff(2 * st + 1)];
      b1.x = lds[ldsIdx + j0 + ldsOff(2 * st + 2)];
      b1.y = lds[ldsIdx + j0 + ldsOff(2 * st + 3)];
      acc0 = __builtin_amdgcn_wmma_f32_16x16x4_f32(
          false, a[st],     false, b0, (short)0, acc0, false, false);
      acc1 = __builtin_amdgcn_wmma_f32_16x16x4_f32(
          false, a[st + 1], false, b1, (short)0, acc1, false, false);
    }
    v8f acc = acc0 + acc1;

    // D: VGPR v = channel v, column = lane (lanes 0..15); pad channels not stored.
    // All store addresses are base + compile-time byte immediate.
    if (hi == 0 && (j0 + p) < Wout) {
      #pragma unroll
      for (int v = 0; v < 8; ++v) {
        __builtin_nontemporal_store(
            acc[v], (float*)(outb + (v * Hout * Wout + j0) * 4));
      }
    }
  }
}

extern "C" void kernel_launch(void* const* d_in, const int* in_sizes, int n_in,
                              void* d_out, int out_size, void* d_ws, size_t ws_size,
                              hipStream_t stream) {
  const float* in  = (const float*)d_in[0];   // (64, 8, 256, 256) f32
  const float* flt = (const float*)d_in[1];   // (8, 8, 3, 3) f32
  float* out = (float*)d_out;                 // (64, 8, 254, 254) f32

  dim3 grid(64 * 32);   // 64 images x 32 row-blocks of 8 rows
  dim3 block(256);      // 8 waves
  conv3x3_wmma_f32<<<grid, block, LDS_BYTES, stream>>>(in, flt, out);
}